// WindowAttention_25056839205752
// MI455X (gfx1250) — compile-verified
//
#include <hip/hip_runtime.h>

#define NPTS 65536
#define CCH  256
#define NH   16
#define HD   16
#define KW   32
#define NW   2048
#define LX   40
#define LR   32

typedef __attribute__((ext_vector_type(2))) float v2f;
typedef __attribute__((ext_vector_type(8))) float v8f;

__device__ __forceinline__ v8f wmma4(v2f a, v2f b, v8f c) {
  // D = A(16x4 f32) * B(4x16 f32) + C(16x16 f32)
  return __builtin_amdgcn_wmma_f32_16x16x4_f32(false, a, false, b, (short)0, c, false, false);
}

// Async global->LDS copy (CDNA5, tracked by ASYNCcnt).  Generic pointers to
// __shared__ objects carry the LDS byte offset in addr[31:0] (ISA 10.2), so a
// 32-bit truncation yields the VDST LDS address the instruction expects.
__device__ __forceinline__ void async_ld_b128(const void* gptr, void* lptr) {
  unsigned lds = (unsigned)(size_t)lptr;
  unsigned long long g = (unsigned long long)(size_t)gptr;
  asm volatile("global_load_async_to_lds_b128 %0, %1, off"
               :: "v"(lds), "v"(g) : "memory");
}
__device__ __forceinline__ void wait_async() {
  asm volatile("s_wait_asynccnt 0x0" ::: "memory");
}

// ---------------------------------------------------------------------------
// Kernel 1: qkv = feats @ qkv_w.T + qkv_b, rows gathered through perm so that
// Q/K/V come out in window order; q scaled by hd^-0.5.
// One wave -> 16x32 output tile (two 16x16 WMMA accumulators per A fragment).
// ---------------------------------------------------------------------------
__global__ __launch_bounds__(256) void qkv_gemm_kernel(
    const float* __restrict__ feats, const float* __restrict__ qkv_w,
    const float* __restrict__ qkv_b, const int* __restrict__ perm,
    float* __restrict__ Qp, float* __restrict__ Kp, float* __restrict__ Vp)
{
  __shared__ float Bs[32][260];          // pitch 260: 16B rows, conflict-free
  const int tid  = threadIdx.x;
  const int wave = tid >> 5, lane = tid & 31;
  const int l16  = lane & 15, hi = lane >> 4;
  const int offK = hi * 2, off8 = hi * 8;
  const int nb = blockIdx.x * 32;
  const int mb = (blockIdx.y * 8 + wave) * 16;

  // async-stage 32x256 f32 slice of qkv_w into LDS: 2048 x b128 chunks
#pragma unroll
  for (int rep = 0; rep < 8; ++rep) {
    int idx = rep * 256 + tid;           // chunk id, 4 floats each
    int col = idx >> 6;                  // 64 chunks per row
    int k4  = (idx & 63) * 4;
    async_ld_b128(qkv_w + (size_t)(nb + col) * CCH + k4, &Bs[col][k4]);
  }
  wait_async();
  __syncthreads();

  const int grow = perm[mb + l16];
  const float* arow = feats + (size_t)grow * CCH;

  v8f acc0 = {}, acc1 = {};
#pragma unroll 4
  for (int kk = 0; kk < 64; ++kk) {
    int k0 = kk * 4 + offK;
    v2f a  = *(const v2f*)(arow + k0);
    v2f b0 = *(const v2f*)(&Bs[l16][k0]);
    v2f b1 = *(const v2f*)(&Bs[16 + l16][k0]);
    acc0 = wmma4(a, b0, acc0);
    acc1 = wmma4(a, b1, acc1);
  }

#pragma unroll
  for (int s = 0; s < 2; ++s) {
    const v8f acc = s ? acc1 : acc0;
    const int col  = nb + s * 16 + l16;
    const int part = col >> 8;           // 0=q, 1=k, 2=v
    const int ch   = col & 255;
    const float bias  = qkv_b[col];
    const float scale = (part == 0) ? 0.25f : 1.0f;   // hd^-0.5
    float* dst = (part == 0) ? Qp : (part == 1) ? Kp : Vp;
#pragma unroll
    for (int r = 0; r < 8; ++r) {
      int row = mb + r + off8;
      dst[(size_t)row * CCH + ch] = (acc[r] + bias) * scale;
    }
  }
}

// ---------------------------------------------------------------------------
// Kernel 2: per-window attention.  Block = 4 waves = 4 heads of one window.
// ---------------------------------------------------------------------------
__global__ __launch_bounds__(128) void attn_kernel(
    const float* __restrict__ Qp, const float* __restrict__ Kp,
    const float* __restrict__ Vp, float* __restrict__ Op,
    const int* __restrict__ perm, const float* __restrict__ n_coords,
    const float* __restrict__ qx, const float* __restrict__ kx, const float* __restrict__ vx,
    const float* __restrict__ qr, const float* __restrict__ kr, const float* __restrict__ vr)
{
  __shared__ float csh[32 * 6];
  __shared__ unsigned char idxS[6 * 32 * 32];
  __shared__ float qsh[4][32 * 20];      // pitch 20: 80B rows, conflict-free
  __shared__ float ksh[4][32 * 20];
  __shared__ float vsh[4][32 * 20];
  __shared__ float scr[4][32 * 34];      // P tiles (pitch 49) then A (pitch 34)

  const int tid  = threadIdx.x;
  const int wave = tid >> 5, lane = tid & 31;
  const int l16  = lane & 15, hi = lane >> 4;
  const int offK = hi * 2, off8 = hi * 8;
  const int w = blockIdx.x;
  const int h = blockIdx.y * 4 + wave;

  float* qs = qsh[wave];
  float* ks = ksh[wave];
  float* vs = vsh[wave];
  float* sc = scr[wave];

  // async-stage this head's q/k/v 32x16 tiles (128 b128 chunks each)
  const float* gq = Qp + (size_t)(w * KW) * CCH + h * HD;
  const float* gk = Kp + (size_t)(w * KW) * CCH + h * HD;
  const float* gv = Vp + (size_t)(w * KW) * CCH + h * HD;
#pragma unroll
  for (int rep = 0; rep < 4; ++rep) {
    int idx = rep * 32 + lane;           // chunk id, 4 floats each
    int i  = idx >> 2;
    int c4 = (idx & 3) * 4;
    async_ld_b128(gq + (size_t)i * CCH + c4, &qs[i * 20 + c4]);
    async_ld_b128(gk + (size_t)i * CCH + c4, &ks[i * 20 + c4]);
    async_ld_b128(gv + (size_t)i * CCH + c4, &vs[i * 20 + c4]);
  }

  if (tid < 192) {                       // scaled window coords
    int i = tid / 6, dch = tid % 6;
    float v = n_coords[(size_t)perm[w * KW + i] * 6 + dch];
    csh[i * 6 + dch] = v * (dch < 3 ? 4.0f : 8.0f);   // QS=4, CQS=8
  }
  __syncthreads();
  for (int t = tid; t < 6144; t += 128) {  // relative-index table (shared by heads)
    int dd = t >> 10, rem = t & 1023, i = rem >> 5, j = rem & 31;
    int Ld = (dd < 3) ? LX : LR;
    int v = (int)floorf(csh[i * 6 + dd] - csh[j * 6 + dd]) + (Ld >> 1);
    v = v < 0 ? 0 : (v > Ld - 1 ? Ld - 1 : v);
    idxS[t] = (unsigned char)v;
  }
  __syncthreads();
  wait_async();                          // q/k/v tiles are wave-local

  v8f zro = {};
  v8f attn[2][2] = {{zro, zro}, {zro, zro}};

  // ---- attn = q @ k^T  (2x2 tiles of 16x16, K=16) ----
#pragma unroll
  for (int kk = 0; kk < 4; ++kk) {
    int k0 = kk * 4 + offK;
    v2f a0 = *(const v2f*)&qs[(0  + l16) * 20 + k0];
    v2f a1 = *(const v2f*)&qs[(16 + l16) * 20 + k0];
    v2f b0 = *(const v2f*)&ks[(0  + l16) * 20 + k0];
    v2f b1 = *(const v2f*)&ks[(16 + l16) * 20 + k0];
    attn[0][0] = wmma4(a0, b0, attn[0][0]);
    attn[0][1] = wmma4(a0, b1, attn[0][1]);
    attn[1][0] = wmma4(a1, b0, attn[1][0]);
    attn[1][1] = wmma4(a1, b1, attn[1][1]);
  }

  // ---- positional bias: P_qk = q @ k_tab^T, P_kq = k @ q_tab^T, then gather ----
  const float* qtabs[2] = {qx, qr};
  const float* ktabs[2] = {kx, kr};
#pragma unroll
  for (int g = 0; g < 2; ++g) {
    const int Ld = g ? LR : LX;
    const int Lt = g ? 2 : 3;
    const float* qt = qtabs[g];
    const float* kt = ktabs[g];
#pragma unroll
    for (int d = 0; d < 3; ++d) {
      const int dd = g * 3 + d;
      // P_qk tiles: rows = i, cols = l
#pragma unroll
      for (int it = 0; it < 2; ++it) {
#pragma unroll
        for (int lt = 0; lt < Lt; ++lt) {
          int l = lt * 16 + l16;
          v8f p = {};
#pragma unroll
          for (int kk = 0; kk < 4; ++kk) {
            int k0 = kk * 4 + offK;
            v2f a = *(const v2f*)&qs[(it * 16 + l16) * 20 + k0];
            v2f b;
            if (l < Ld) b = *(const v2f*)&kt[((size_t)(d * Ld + l) * NH + h) * HD + k0];
            else { b.x = 0.f; b.y = 0.f; }
            p = wmma4(a, b, p);
          }
#pragma unroll
          for (int r = 0; r < 8; ++r) sc[(r + off8) * 49 + lt * 16 + l16] = p[r];
        }
#pragma unroll
        for (int jt = 0; jt < 2; ++jt)
#pragma unroll
          for (int r = 0; r < 8; ++r) {
            int rowi = it * 16 + r + off8;
            int j = jt * 16 + l16;
            int idx = idxS[dd * 1024 + rowi * 32 + j];
            attn[it][jt][r] += sc[(r + off8) * 49 + idx];
          }
      }
      // P_kq tiles: rows = j, cols = l
#pragma unroll
      for (int jt = 0; jt < 2; ++jt) {
#pragma unroll
        for (int lt = 0; lt < Lt; ++lt) {
          int l = lt * 16 + l16;
          v8f p = {};
#pragma unroll
          for (int kk = 0; kk < 4; ++kk) {
            int k0 = kk * 4 + offK;
            v2f a = *(const v2f*)&ks[(jt * 16 + l16) * 20 + k0];
            v2f b;
            if (l < Ld) b = *(const v2f*)&qt[((size_t)(d * Ld + l) * NH + h) * HD + k0];
            else { b.x = 0.f; b.y = 0.f; }
            p = wmma4(a, b, p);
          }
#pragma unroll
          for (int r = 0; r < 8; ++r) sc[(r + off8) * 49 + lt * 16 + l16] = p[r];
        }
#pragma unroll
        for (int it = 0; it < 2; ++it)
#pragma unroll
          for (int r = 0; r < 8; ++r) {
            int rowi = it * 16 + r + off8;
            int j = jt * 16 + l16;
            int idx = idxS[dd * 1024 + rowi * 32 + j];
            attn[it][jt][r] += sc[l16 * 49 + idx];
          }
      }
    }
  }

  // ---- softmax over j (rows live in 16-lane groups) + write A to LDS ----
#pragma unroll
  for (int it = 0; it < 2; ++it)
#pragma unroll
    for (int r = 0; r < 8; ++r) {
      float m = fmaxf(attn[it][0][r], attn[it][1][r]);
#pragma unroll
      for (int s = 8; s >= 1; s >>= 1) m = fmaxf(m, __shfl_xor(m, s, 32));
      float e0 = __expf(attn[it][0][r] - m);
      float e1 = __expf(attn[it][1][r] - m);
      float sm = e0 + e1;
#pragma unroll
      for (int s = 8; s >= 1; s >>= 1) sm += __shfl_xor(sm, s, 32);
      float rinv = 1.0f / sm;
      int rowi = it * 16 + r + off8;
      sc[rowi * 34 + l16]      = e0 * rinv;
      sc[rowi * 34 + 16 + l16] = e1 * rinv;
    }

  // ---- out = A @ V  (V from LDS) ----
  v8f outA[2] = {zro, zro};
#pragma unroll
  for (int kk = 0; kk < 8; ++kk) {
    int k0 = kk * 4 + offK;
    v2f b;
    b.x = vs[(k0)     * 20 + l16];
    b.y = vs[(k0 + 1) * 20 + l16];
    v2f a0 = *(const v2f*)&sc[(0  + l16) * 34 + k0];
    v2f a1 = *(const v2f*)&sc[(16 + l16) * 34 + k0];
    outA[0] = wmma4(a0, b, outA[0]);
    outA[1] = wmma4(a1, b, outA[1]);
  }

  // ---- value positional tables: out[i,c] += sum_j a[i,j] * v_tab[idx(i,j)][h][c] ----
  const float* vtabs[2] = {vx, vr};
  for (int g = 0; g < 2; ++g) {
    const int Ld = g ? LR : LX;
    const float* vt = vtabs[g];
    for (int d = 0; d < 3; ++d) {
      const int dd = g * 3 + d;
#pragma unroll
      for (int it = 0; it < 2; ++it)
#pragma unroll
        for (int r = 0; r < 8; ++r) {
          int rowi = it * 16 + r + off8;
          float acc = 0.f;
          for (int j = 0; j < 32; ++j) {
            float aij = sc[rowi * 34 + j];
            int idx = idxS[dd * 1024 + rowi * 32 + j];
            acc += aij * vt[((size_t)(d * Ld + idx) * NH + h) * HD + l16];
          }
          outA[it][r] += acc;
        }
    }
  }

#pragma unroll
  for (int it = 0; it < 2; ++it)
#pragma unroll
    for (int r = 0; r < 8; ++r) {
      int rowi = it * 16 + r + off8;
      Op[(size_t)(w * KW + rowi) * CCH + h * HD + l16] = outA[it][r];
    }
}

// ---------------------------------------------------------------------------
// Kernel 3: out[perm[r], :] = Op[r, :] @ proj_w.T + proj_b  (scatter fused)
// ---------------------------------------------------------------------------
__global__ __launch_bounds__(256) void proj_gemm_kernel(
    const float* __restrict__ Op, const float* __restrict__ proj_w,
    const float* __restrict__ proj_b, const int* __restrict__ perm,
    float* __restrict__ out)
{
  __shared__ float Bs[32][260];
  const int tid  = threadIdx.x;
  const int wave = tid >> 5, lane = tid & 31;
  const int l16  = lane & 15, hi = lane >> 4;
  const int offK = hi * 2, off8 = hi * 8;
  const int nb = blockIdx.x * 32;
  const int mb = (blockIdx.y * 8 + wave) * 16;

#pragma unroll
  for (int rep = 0; rep < 8; ++rep) {
    int idx = rep * 256 + tid;
    int col = idx >> 6;
    int k4  = (idx & 63) * 4;
    async_ld_b128(proj_w + (size_t)(nb + col) * CCH + k4, &Bs[col][k4]);
  }
  wait_async();
  __syncthreads();

  const float* arow = Op + (size_t)(mb + l16) * CCH;

  v8f acc0 = {}, acc1 = {};
#pragma unroll 4
  for (int kk = 0; kk < 64; ++kk) {
    int k0 = kk * 4 + offK;
    v2f a  = *(const v2f*)(arow + k0);
    v2f b0 = *(const v2f*)(&Bs[l16][k0]);
    v2f b1 = *(const v2f*)(&Bs[16 + l16][k0]);
    acc0 = wmma4(a, b0, acc0);
    acc1 = wmma4(a, b1, acc1);
  }

#pragma unroll
  for (int s = 0; s < 2; ++s) {
    const v8f acc = s ? acc1 : acc0;
    const int col = nb + s * 16 + l16;
    const float bias = proj_b[col];
#pragma unroll
    for (int r = 0; r < 8; ++r) {
      int row  = mb + r + off8;
      int prow = perm[row];
      out[(size_t)prow * CCH + col] = acc[r] + bias;
    }
  }
}

extern "C" void kernel_launch(void* const* d_in, const int* in_sizes, int n_in,
                              void* d_out, int out_size, void* d_ws, size_t ws_size,
                              hipStream_t stream) {
  const float* feats    = (const float*)d_in[0];
  const float* n_coords = (const float*)d_in[1];
  const int*   perm     = (const int*)d_in[2];
  const float* qx = (const float*)d_in[3];
  const float* kx = (const float*)d_in[4];
  const float* vx = (const float*)d_in[5];
  const float* qr = (const float*)d_in[6];
  const float* kr = (const float*)d_in[7];
  const float* vr = (const float*)d_in[8];
  const float* qkv_w  = (const float*)d_in[9];
  const float* qkv_b  = (const float*)d_in[10];
  const float* proj_w = (const float*)d_in[11];
  const float* proj_b = (const float*)d_in[12];
  float* out = (float*)d_out;

  float* Qp = (float*)d_ws;
  float* Kp = Qp + (size_t)NPTS * CCH;
  float* Vp = Kp + (size_t)NPTS * CCH;
  float* Op = Vp + (size_t)NPTS * CCH;

  qkv_gemm_kernel<<<dim3(24, 512), 256, 0, stream>>>(feats, qkv_w, qkv_b, perm, Qp, Kp, Vp);
  attn_kernel<<<dim3(NW, 4), 128, 0, stream>>>(Qp, Kp, Vp, Op, perm, n_coords,
                                               qx, kx, vx, qr, kr, vr);
  proj_gemm_kernel<<<dim3(8, 512), 256, 0, stream>>>(Op, proj_w, proj_b, perm, out);
}